// LNon_37460704756094
// MI455X (gfx1250) — compile-verified
//
#include <hip/hip_runtime.h>
#include <math.h>

typedef __attribute__((ext_vector_type(2))) float v2f;
typedef __attribute__((ext_vector_type(4))) float v4f;
typedef __attribute__((ext_vector_type(8))) float v8f;
typedef __attribute__((ext_vector_type(4))) int   v4i;

#define AS1 __attribute__((address_space(1)))
#define AS3 __attribute__((address_space(3)))

#define NBLK 1024
#define NTHR 256
#define POINTS 120

#if defined(__gfx1250__) && __has_builtin(__builtin_amdgcn_wmma_f32_16x16x4_f32)
#define WMMA_OK 1
#else
#define WMMA_OK 0
#endif

#if defined(__gfx1250__) && __has_builtin(__builtin_amdgcn_global_load_async_to_lds_b128) && \
    __has_builtin(__builtin_amdgcn_s_wait_asynccnt)
#define ASYNC_OK 1
#else
#define ASYNC_OK 0
#endif

// ---------------------------------------------------------------- helpers ---

__device__ __forceinline__ float lut_interp(const float* lut, float idx) {
  float b = floorf(idx);
  b = fminf(fmaxf(b, 0.f), (float)(POINTS - 1));
  int bi = (int)b;
  float pos = idx - b;
  int ei = bi + 1; if (ei > POINTS - 1) ei = POINTS - 1;
  return (1.f - pos) * lut[bi] + pos * lut[ei];
}

__device__ __forceinline__ float foilize(float x, const float* lutT, const float* lutV) {
  float sig = 1.f / (1.f + expf(-x));
  float th  = lut_interp(lutT, sig * (float)(POINTS - 1));
  float at  = fabsf(tanhf(x));
  float ve  = lut_interp(lutV, at * (float)(POINTS - 1));
  float sth, cth;
  sincosf(th, &sth, &cth);
  return x * expf(ve * sth) + ve * cth;
}

// Wave32 reduction of (s, q) using one V_WMMA_F32_16X16X4_F32:
// A[m][k] holds the 32 lane partials (a[0]=s, a[1]=q per the 16x4 f32 layout);
// B selects K={0,2} -> column 0 (sum of s) and K={1,3} -> column 1 (sum of q).
__device__ __forceinline__ void wave_reduce_sq(float& s, float& q) {
#if WMMA_OK
  const int lane = (int)(threadIdx.x & 31);
  v2f a; a[0] = s; a[1] = q;
  v2f b;
  b[0] = (lane == 0 || lane == 16) ? 1.f : 0.f;   // B[0][0], B[2][0]
  b[1] = (lane == 1 || lane == 17) ? 1.f : 0.f;   // B[1][1], B[3][1]
  v8f c = {};
  v8f d = __builtin_amdgcn_wmma_f32_16x16x4_f32(false, a, false, b, (short)0, c, false, false);
  float t = ((d[0] + d[1]) + (d[2] + d[3])) + ((d[4] + d[5]) + (d[6] + d[7]));
  s = __shfl(t, 0, 32) + __shfl(t, 16, 32);   // rows 0..7 + rows 8..15 of column 0
  q = __shfl(t, 1, 32) + __shfl(t, 17, 32);   // column 1
#else
  for (int off = 16; off > 0; off >>= 1) {
    s += __shfl_xor(s, off, 32);
    q += __shfl_xor(q, off, 32);
  }
#endif
}

__device__ __forceinline__ void block_store_partial(float s, float q, float* parts) {
  __shared__ float wS[NTHR / 32], wQ[NTHR / 32];
  wave_reduce_sq(s, q);
  int lane = (int)(threadIdx.x & 31), w = (int)(threadIdx.x >> 5);
  if (lane == 0) { wS[w] = s; wQ[w] = q; }
  __syncthreads();
  if (threadIdx.x == 0) {
    float S = 0.f, Q = 0.f;
#pragma unroll
    for (int k = 0; k < NTHR / 32; ++k) { S += wS[k]; Q += wQ[k]; }
    parts[2 * blockIdx.x]     = S;
    parts[2 * blockIdx.x + 1] = Q;
  }
}

#if ASYNC_OK
__device__ __forceinline__ void async_copy16(const v4f* g, v4f* l) {
  // signature (probe-confirmed): (v4i AS1* src, v4i AS3* dst, imm offset, imm cpol)
  __builtin_amdgcn_global_load_async_to_lds_b128(
      (AS1 v4i*)(void*)g, (AS3 v4i*)(void*)l, 0, 0);
}
#endif

// ---------------------------------------------------------------- kernels ---

// Pass 1: sum / sumsq of raw data. Streams HBM->L2 via async global->LDS DMA
// (double buffered), leaving the tensor resident in the 192MB L2 for passes 2/3.
__global__ __launch_bounds__(NTHR) void k_stats_data(const float* __restrict__ data,
                                                     long long n,
                                                     float* __restrict__ parts) {
  const v4f* dv = (const v4f*)data;
  long long n4 = n >> 2;
  long long stride = (long long)gridDim.x * NTHR;
  long long idx = (long long)blockIdx.x * NTHR + threadIdx.x;
  float s = 0.f, q = 0.f;
#if ASYNC_OK
  __shared__ v4f stage[2][NTHR];
  int buf = 0;
  bool have = idx < n4;
  if (have) async_copy16(dv + idx, &stage[0][threadIdx.x]);
  while (have) {
    long long nidx = idx + stride;
    bool hn = nidx < n4;
    if (hn) {
      async_copy16(dv + nidx, &stage[buf ^ 1][threadIdx.x]);
      __builtin_amdgcn_s_wait_asynccnt(1);   // oldest (ours) has landed in LDS
    } else {
      __builtin_amdgcn_s_wait_asynccnt(0);
    }
    asm volatile("" ::: "memory");
    v4f v = stage[buf][threadIdx.x];
    s += (v[0] + v[1]) + (v[2] + v[3]);
    q += (v[0] * v[0] + v[1] * v[1]) + (v[2] * v[2] + v[3] * v[3]);
    buf ^= 1; idx = nidx; have = hn;
  }
#else
  for (; idx < n4; idx += stride) {
    v4f v = dv[idx];
    s += (v[0] + v[1]) + (v[2] + v[3]);
    q += (v[0] * v[0] + v[1] * v[1]) + (v[2] * v[2] + v[3] * v[3]);
  }
#endif
  long long tb = n4 << 2;
  if (blockIdx.x == 0 && (long long)threadIdx.x < (n - tb)) {
    float v = data[tb + threadIdx.x];
    s += v; q += v * v;
  }
  block_store_partial(s, q, parts);
}

// Deterministic final reduction of per-block partials -> (mean, rstd*scale).
__global__ __launch_bounds__(NTHR) void k_reduce(const float* __restrict__ parts, int nparts,
                                                 const float* __restrict__ scale,
                                                 double n, float* __restrict__ outMS) {
  __shared__ double sS[NTHR], sQ[NTHR];
  double s = 0.0, q = 0.0;
  for (int i = threadIdx.x; i < nparts; i += NTHR) {
    s += (double)parts[2 * i];
    q += (double)parts[2 * i + 1];
  }
  sS[threadIdx.x] = s; sQ[threadIdx.x] = q;
  __syncthreads();
  for (int off = NTHR / 2; off > 0; off >>= 1) {
    if ((int)threadIdx.x < off) {
      sS[threadIdx.x] += sS[threadIdx.x + off];
      sQ[threadIdx.x] += sQ[threadIdx.x + off];
    }
    __syncthreads();
  }
  if (threadIdx.x == 0) {
    double S = sS[0], Q = sQ[0];
    double mean = S / n;
    double var = (Q - S * S / n) / (n - 1.0);   // ddof = 1
    if (var < 0.0) var = 0.0;
    outMS[0] = (float)mean;
    outMS[1] = (float)(1.0 / sqrt(var)) * scale[0];
  }
}

// Pass 2: sum / sumsq of foilize((d-mean)*rstd*scalei). Re-reads data (L2 hit),
// does NOT materialize val (recompute in pass 3 is cheaper than 2x134MB traffic).
__global__ __launch_bounds__(NTHR) void k_stats_val(const float* __restrict__ data,
                                                    const float* __restrict__ params,
                                                    long long n,
                                                    const float* __restrict__ stats,
                                                    float* __restrict__ parts) {
  __shared__ float lutT[POINTS], lutV[POINTS];
  if (threadIdx.x < POINTS) {
    lutT[threadIdx.x] = params[threadIdx.x];
    lutV[threadIdx.x] = params[POINTS + threadIdx.x];
  }
  __syncthreads();
  const float meanD = stats[0], rsD = stats[1];
  const v4f* dv = (const v4f*)data;
  long long n4 = n >> 2;
  long long stride = (long long)gridDim.x * NTHR;
  float s = 0.f, q = 0.f;
  for (long long idx = (long long)blockIdx.x * NTHR + threadIdx.x; idx < n4; idx += stride) {
    __builtin_prefetch(dv + idx + stride, 0, 0);
    v4f v = dv[idx];
#pragma unroll
    for (int c = 0; c < 4; ++c) {
      float x = (v[c] - meanD) * rsD;
      float val = foilize(x, lutT, lutV);
      s += val; q += val * val;
    }
  }
  long long tb = n4 << 2;
  if (blockIdx.x == 0 && (long long)threadIdx.x < (n - tb)) {
    float x = (data[tb + threadIdx.x] - meanD) * rsD;
    float val = foilize(x, lutT, lutV);
    s += val; q += val * val;
  }
  block_store_partial(s, q, parts);
}

// Pass 3: recompute val, normalize with val stats, non-temporal store of output.
__global__ __launch_bounds__(NTHR) void k_final(const float* __restrict__ data,
                                                const float* __restrict__ params,
                                                long long n,
                                                const float* __restrict__ stats,
                                                float* __restrict__ out) {
  __shared__ float lutT[POINTS], lutV[POINTS];
  if (threadIdx.x < POINTS) {
    lutT[threadIdx.x] = params[threadIdx.x];
    lutV[threadIdx.x] = params[POINTS + threadIdx.x];
  }
  __syncthreads();
  const float meanD = stats[0], rsD = stats[1];
  const float meanV = stats[2], rsV = stats[3];
  const v4f* dv = (const v4f*)data;
  v4f* ov = (v4f*)out;
  long long n4 = n >> 2;
  long long stride = (long long)gridDim.x * NTHR;
  for (long long idx = (long long)blockIdx.x * NTHR + threadIdx.x; idx < n4; idx += stride) {
    __builtin_prefetch(dv + idx + stride, 0, 0);
    v4f v = dv[idx];
    v4f o;
#pragma unroll
    for (int c = 0; c < 4; ++c) {
      float x = (v[c] - meanD) * rsD;
      o[c] = (foilize(x, lutT, lutV) - meanV) * rsV;
    }
    __builtin_nontemporal_store(o, ov + idx);   // don't pollute L2 with the output
  }
  long long tb = n4 << 2;
  if (blockIdx.x == 0 && (long long)threadIdx.x < (n - tb)) {
    long long i = tb + threadIdx.x;
    float x = (data[i] - meanD) * rsD;
    out[i] = (foilize(x, lutT, lutV) - meanV) * rsV;
  }
}

// ---------------------------------------------------------------- launch ----

extern "C" void kernel_launch(void* const* d_in, const int* in_sizes, int n_in,
                              void* d_out, int out_size, void* d_ws, size_t ws_size,
                              hipStream_t stream) {
  const float* data   = (const float*)d_in[0];
  const float* params = (const float*)d_in[1];
  const float* scalei = (const float*)d_in[2];
  const float* scaleo = (const float*)d_in[3];
  float* out = (float*)d_out;
  float* wsf = (float*)d_ws;

  long long n = (long long)in_sizes[0];
  double nd = (double)n;

  // ws layout (floats): [0..3] final stats (meanD, rstdD*scalei, meanV, rstdV*scaleo)
  //                     [16 .. 16+2*NBLK)         pass-1 per-block partials
  //                     [16+2*NBLK .. 16+4*NBLK)  pass-2 per-block partials
  float* partsA = wsf + 16;
  float* partsB = wsf + 16 + 2 * NBLK;

  k_stats_data<<<NBLK, NTHR, 0, stream>>>(data, n, partsA);
  k_reduce<<<1, NTHR, 0, stream>>>(partsA, NBLK, scalei, nd, wsf + 0);
  k_stats_val<<<NBLK, NTHR, 0, stream>>>(data, params, n, wsf, partsB);
  k_reduce<<<1, NTHR, 0, stream>>>(partsB, NBLK, scaleo, nd, wsf + 2);
  k_final<<<NBLK, NTHR, 0, stream>>>(data, params, n, wsf, out);
}